// LSTM_58351425684032
// MI455X (gfx1250) — compile-verified
//
#include <hip/hip_runtime.h>

#define T_STEPS 512
#define BATCH   4096
#define HID     12
#define G4      48      // 4*HID
#define TILE_B  16

typedef __attribute__((ext_vector_type(2))) float v2f;
typedef __attribute__((ext_vector_type(8))) float v8f;

// Branch-free tanh: native V_TANH_F32 on CDNA5 if the builtin exists,
// otherwise exp2+rcp (saturates correctly at +-1, no NaN for finite x).
__device__ __forceinline__ float fast_tanh(float x) {
#if __has_builtin(__builtin_amdgcn_tanhf)
    return __builtin_amdgcn_tanhf(x);
#else
    const float e = __builtin_amdgcn_exp2f(x * 2.8853900817779268f); // exp(2x)
    return 1.0f - 2.0f * __builtin_amdgcn_rcpf(1.0f + e);
#endif
}

__device__ __forceinline__ float fast_sigmoid(float x) {
    return fmaf(0.5f, fast_tanh(0.5f * x), 0.5f);
}

__global__ __launch_bounds__(32)
void lstm_wmma_f32_kernel(const float* __restrict__ X,
                          const float* __restrict__ W_ih,
                          const float* __restrict__ W_hh,
                          const float* __restrict__ b_ih,
                          const float* __restrict__ b_hh,
                          const float* __restrict__ fc_w,
                          const float* __restrict__ fc_b,
                          float* __restrict__ out)
{
    __shared__ float lds_gates[TILE_B * G4];   // [m][gate]
    __shared__ float lds_h[TILE_B * HID];      // [m][j]

    const int lane = threadIdx.x;     // 0..31 (wave32)
    const int half = lane >> 4;       // 0 = lanes 0-15, 1 = lanes 16-31
    const int l16  = lane & 15;
    const int b0   = blockIdx.x * TILE_B;

    // ---- Constant B fragments: augmented [W_hh^T ; W_ih ; b_ih+b_hh ; 0 ; 0]
    // gates(16x16 tile n) = A(16x16) x B(16x16) as 4 chunks of K=4.
    // B 4x16 f32 layout (mirror of documented A 16x4):
    //   v0: K = 4c + 2*half, v1: K = 4c + 2*half + 1, N = l16
    v2f Bfrag[3][4];
#pragma unroll
    for (int n = 0; n < 3; ++n) {
        const int gate = 16 * n + l16;
#pragma unroll
        for (int c = 0; c < 3; ++c) {          // K = 0..11 : W_hh^T
            const int j0 = 4 * c + 2 * half;
            Bfrag[n][c].x = W_hh[gate * HID + j0];
            Bfrag[n][c].y = W_hh[gate * HID + j0 + 1];
        }
        // chunk 3: K=12 -> W_ih (IN==1), K=13 -> fused bias, K=14,15 -> 0
        Bfrag[n][3].x = half ? 0.f : W_ih[gate];
        Bfrag[n][3].y = half ? 0.f : (b_ih[gate] + b_hh[gate]);
    }

    // activation phase: lane owns 6 static (m,j) pairs: p = lane*6 + q
    float fcw_q[6];
    int mq[6], jq[6];
#pragma unroll
    for (int q = 0; q < 6; ++q) {
        const int p = lane * 6 + q;
        mq[q] = p / HID;
        jq[q] = p % HID;
        fcw_q[q] = fc_w[jq[q]];
    }
    const float fcb = fc_b[0];

    // recurrent state
    float cst[6];
#pragma unroll
    for (int q = 0; q < 6; ++q) cst[q] = 0.f;

    // A fragments: [h(K=0..11) | x(K=12) | 1(K=13) | 0 | 0], M = l16
    v2f Afrag[4];
#pragma unroll
    for (int c = 0; c < 3; ++c) { Afrag[c].x = 0.f; Afrag[c].y = 0.f; }
    Afrag[3].x = 0.f;
    Afrag[3].y = half ? 0.f : 1.0f;       // constant: 1 feeds the bias column

    const float* Xp = X + b0;
    float xv = Xp[l16];                   // software-pipelined X (step 0)

    for (int t = 0; t < T_STEPS; ++t) {
        // X for THIS step arrived a full iteration ago -> register-ready
        Afrag[3].x = half ? 0.f : xv;

        // issue next step's X load + prefetch of the row after (clamped:
        // WGP-scope prefetch is non-speculative, keep addresses in-bounds)
        const float* nxt = Xp + ((t + 1 < T_STEPS) ? BATCH : 0);
        const float xv_next = nxt[l16];
        __builtin_prefetch(Xp + ((t + 2 < T_STEPS) ? 2 * BATCH : 0), 0, 3);
        Xp += BATCH;

        // gates = [h|x|1] @ [W_hh^T ; W_ih ; bias]   (C starts at inline 0)
        // augmented chunk first: its A operand is register-ready, so the
        // h-fragment ds_load waits overlap with WMMA issue.
        v8f D[3];
#pragma unroll
        for (int n = 0; n < 3; ++n) {
            v8f acc = {};
            acc = __builtin_amdgcn_wmma_f32_16x16x4_f32(
                    false, Afrag[3], false, Bfrag[n][3], (short)0, acc,
                    false, false);
#pragma unroll
            for (int kc = 0; kc < 3; ++kc)
                acc = __builtin_amdgcn_wmma_f32_16x16x4_f32(
                        false, Afrag[kc], false, Bfrag[n][kc], (short)0, acc,
                        false, false);
            D[n] = acc;
        }

        // scatter gates to LDS: m = r + 8*half, gate = 16n + l16
#pragma unroll
        for (int n = 0; n < 3; ++n) {
            const int gate = 16 * n + l16;
#pragma unroll
            for (int r = 0; r < 8; ++r)
                lds_gates[(r + 8 * half) * G4 + gate] = D[n][r];
        }
        __syncthreads();   // single-wave WG: zero-cost scheduling fence

        // activations + cell/hidden update + fused FC partial
        float partial = 0.f;
#pragma unroll
        for (int q = 0; q < 6; ++q) {
            const float* gm = &lds_gates[mq[q] * G4];
            const float gi = gm[jq[q]];
            const float gf = gm[HID + jq[q]];
            const float gg = gm[2 * HID + jq[q]];
            const float go = gm[3 * HID + jq[q]];
            const float i_ = fast_sigmoid(gi);
            const float f_ = fast_sigmoid(gf);
            const float g_ = fast_tanh(gg);
            const float o_ = fast_sigmoid(go);
            const float cn = fmaf(f_, cst[q], i_ * g_);
            cst[q] = cn;
            const float hn = o_ * fast_tanh(cn);
            lds_h[mq[q] * HID + jq[q]] = hn;
            partial = fmaf(hn, fcw_q[q], partial);
        }
        // lanes (2m, 2m+1) hold the two halves of batch row m's FC dot product
        partial += __shfl_xor(partial, 1, 32);
        if ((lane & 1) == 0)
            out[t * BATCH + b0 + (lane >> 1)] = partial + fcb;
        __syncthreads();

        // rebuild A fragments (h in 16x4 f32 A layout): M=l16, K = 4kc+2*half+{0,1}
#pragma unroll
        for (int kc = 0; kc < 3; ++kc) {
            const int j0 = 4 * kc + 2 * half;
            Afrag[kc].x = lds_h[l16 * HID + j0];
            Afrag[kc].y = lds_h[l16 * HID + j0 + 1];
        }
        __syncthreads();

        xv = xv_next;
    }
}

extern "C" void kernel_launch(void* const* d_in, const int* in_sizes, int n_in,
                              void* d_out, int out_size, void* d_ws, size_t ws_size,
                              hipStream_t stream) {
    (void)in_sizes; (void)n_in; (void)d_ws; (void)ws_size; (void)out_size;
    const float* X    = (const float*)d_in[0];
    const float* W_ih = (const float*)d_in[1];
    const float* W_hh = (const float*)d_in[2];
    const float* b_ih = (const float*)d_in[3];
    const float* b_hh = (const float*)d_in[4];
    const float* fc_w = (const float*)d_in[5];
    const float* fc_b = (const float*)d_in[6];
    float* out = (float*)d_out;

    dim3 grid(BATCH / TILE_B);   // 256 independent batch tiles, full scan each
    dim3 block(32);              // one wave32 per tile
    lstm_wmma_f32_kernel<<<grid, block, 0, stream>>>(X, W_ih, W_hh, b_ih, b_hh,
                                                     fc_w, fc_b, out);
}